// pre_GCN_82411832475695
// MI455X (gfx1250) — compile-verified
//
#include <hip/hip_runtime.h>
#include <hip/hip_bf16.h>

typedef __attribute__((ext_vector_type(2))) float v2f;
typedef __attribute__((ext_vector_type(8))) float v8f;

#define TM 128
#define TN 128
#define TK 16

// ---------------------------------------------------------------------------
// Generic fp32 WMMA GEMM: C[M,N] = A[M,K] @ (BT ? B[N,K]^T : B[K,N])  (+epilogue)
// EPI: 0 = none, 1 = +bias, 2 = relu(+bias), 3 = leaky-relu(0.05), no bias
// Block: 256 threads = 8 waves; block tile 128x128; wave tile 32x64 (2x4 WMMA).
// LDS tiles are stored K-pair interleaved: element [kp][m] = (X[m][2kp], X[m][2kp+1])
// so each WMMA fragment is one aligned ds_load_b64 into an even VGPR pair.
// Software pipeline: next chunk's global loads are issued while WMMAs run.
// K must be a multiple of 16; M,N multiples of 128 (true for all shapes here).
// ---------------------------------------------------------------------------
template<int EPI, bool BT>
__global__ __launch_bounds__(256)
void wmma_gemm_f32(const float* __restrict__ Ag, const float* __restrict__ Bg,
                   const float* __restrict__ bias, float* __restrict__ Cg,
                   int M, int N, int K, long sA, long sB, long sC)
{
    constexpr int KP = TK / 2;          // 8 k-pairs per chunk
    constexpr int APITCH = TM + 16;     // v2f units; 288 floats (mod 64 = 32)
    constexpr int BPITCH = TN + 16;
    __shared__ v2f Asp[KP][APITCH];
    __shared__ v2f Bsp[KP][BPITCH];

    const int batch = blockIdx.z;
    Ag += (long)batch * sA;
    Bg += (long)batch * sB;
    Cg += (long)batch * sC;

    const int tid  = threadIdx.x;
    const int wave = tid >> 5;          // 0..7
    const int lane = tid & 31;
    const int half = lane >> 4;         // 0: lanes 0-15, 1: lanes 16-31
    const int r    = lane & 15;
    const int wrow = (wave >> 1) * 32;  // 0,32,64,96
    const int wcol = (wave & 1) * 64;   // 0,64

    const int m0 = blockIdx.y * TM;
    const int n0 = blockIdx.x * TN;

    const int lda = K;
    const int ldb = BT ? K : N;

    const v8f vzero = {0.f,0.f,0.f,0.f,0.f,0.f,0.f,0.f};
    v8f acc[2][4];
#pragma unroll
    for (int i = 0; i < 2; ++i)
#pragma unroll
        for (int j = 0; j < 4; ++j) acc[i][j] = vzero;

    v2f aReg[4], bReg[4];   // register staging for the software pipeline

    auto ldA = [&](int k0) {
#pragma unroll
        for (int i = 0; i < 4; ++i) {
            const int idx = tid + 256 * i;      // 1024 k-pairs
            const int kp = idx & (KP - 1);
            const int mm = idx >> 3;
            aReg[i] = *(const v2f*)(Ag + (long)(m0 + mm) * lda + k0 + 2 * kp);
        }
    };
    auto ldB = [&](int k0) {
        if (BT) {
#pragma unroll
            for (int i = 0; i < 4; ++i) {
                const int idx = tid + 256 * i;
                const int kp = idx & (KP - 1);
                const int nn = idx >> 3;
                bReg[i] = *(const v2f*)(Bg + (long)(n0 + nn) * ldb + k0 + 2 * kp);
            }
        } else {
#pragma unroll
            for (int i = 0; i < 4; ++i) {
                const int idx = tid + 256 * i;
                const int nn = idx & (TN - 1);
                const int kp = idx >> 7;
                v2f t;
                t.x = Bg[(long)(k0 + 2 * kp    ) * ldb + n0 + nn];
                t.y = Bg[(long)(k0 + 2 * kp + 1) * ldb + n0 + nn];
                bReg[i] = t;
            }
        }
    };
    auto stTiles = [&]() {
#pragma unroll
        for (int i = 0; i < 4; ++i) {
            const int idx = tid + 256 * i;
            { const int kp = idx & (KP - 1); const int mm = idx >> 3; Asp[kp][mm] = aReg[i]; }
            if (BT) {
                const int kp = idx & (KP - 1); const int nn = idx >> 3;
                Bsp[kp][nn] = bReg[i];
            } else {
                const int nn = idx & (TN - 1); const int kp = idx >> 7;
                Bsp[kp][nn] = bReg[i];
            }
        }
    };

    ldA(0);
    ldB(0);

    for (int k0 = 0; k0 < K; k0 += TK) {
        stTiles();
        __syncthreads();

        // issue next chunk's global loads; they complete while WMMAs run
        if (k0 + TK < K) { ldA(k0 + TK); ldB(k0 + TK); }

        // 4 WMMA k-steps of K=4. ISA 16x4 fp32 fragment layout:
        // lanes 0-15 hold K={4s,4s+1}, lanes 16-31 hold K={4s+2,4s+3}
        // -> pair index kp = 2s + half; one ds_load_b64 per fragment.
#pragma unroll
        for (int s = 0; s < 4; ++s) {
            const int kp = 2 * s + half;
            v2f afr[2], bfr[4];
            afr[0] = Asp[kp][wrow + r];
            afr[1] = Asp[kp][wrow + 16 + r];
#pragma unroll
            for (int j = 0; j < 4; ++j)
                bfr[j] = Bsp[kp][wcol + 16 * j + r];
#pragma unroll
            for (int i = 0; i < 2; ++i)
#pragma unroll
                for (int j = 0; j < 4; ++j)
                    acc[i][j] = __builtin_amdgcn_wmma_f32_16x16x4_f32(
                        false, afr[i], false, bfr[j],
                        (short)0, acc[i][j], false, false);
        }
        __syncthreads();
    }

    // Epilogue. C/D layout: VGPR v -> row (v + 8*half), lane r -> col r.
#pragma unroll
    for (int i = 0; i < 2; ++i) {
#pragma unroll
        for (int j = 0; j < 4; ++j) {
            const int col = n0 + wcol + 16 * j + r;
            float bv = 0.f;
            if (EPI == 1 || EPI == 2) bv = bias[col];
#pragma unroll
            for (int v = 0; v < 8; ++v) {
                const int row = m0 + wrow + 16 * i + 8 * half + v;
                float cv = acc[i][j][v] + bv;
                if (EPI == 2) cv = fmaxf(cv, 0.f);
                if (EPI == 3) cv = (cv >= 0.f) ? cv : 0.05f * cv;
                Cg[(long)row * N + col] = cv;
            }
        }
    }
}

// ---------------------------------------------------------------------------
// Deterministic two-stage mean over each [L,L] sample of Q
// ---------------------------------------------------------------------------
__global__ __launch_bounds__(256)
void q_sum_stage1(const float* __restrict__ Q, float* __restrict__ part,
                  long perSample, int blocksPer)
{
    const int n = blockIdx.y;
    const long base = (long)n * perSample;
    const long start = (long)blockIdx.x * 256 * 16 + threadIdx.x;
    float s = 0.f;
#pragma unroll
    for (int i = 0; i < 16; ++i) {
        long off = start + (long)i * 256;
        if (off < perSample) s += Q[base + off];
    }
    __shared__ float red[256];
    red[threadIdx.x] = s;
    __syncthreads();
    for (int st = 128; st > 0; st >>= 1) {
        if (threadIdx.x < st) red[threadIdx.x] += red[threadIdx.x + st];
        __syncthreads();
    }
    if (threadIdx.x == 0) part[(long)n * blocksPer + blockIdx.x] = red[0];
}

__global__ __launch_bounds__(256)
void q_sum_stage2(const float* __restrict__ part, float* __restrict__ mean,
                  int blocksPer, float invCount)
{
    const int n = blockIdx.x;
    float s = 0.f;
    for (int i = threadIdx.x; i < blocksPer; i += 256)
        s += part[(long)n * blocksPer + i];
    __shared__ float red[256];
    red[threadIdx.x] = s;
    __syncthreads();
    for (int st = 128; st > 0; st >>= 1) {
        if (threadIdx.x < st) red[threadIdx.x] += red[threadIdx.x + st];
        __syncthreads();
    }
    if (threadIdx.x == 0) mean[n] = red[0] * invCount;
}

// Q = relu(Q - mean[n]), vectorized float4
__global__ __launch_bounds__(256)
void q_relu_sub_mean(float* __restrict__ Q, const float* __restrict__ mean,
                     long perSample4)
{
    const int n = blockIdx.y;
    const float mu = mean[n];
    float4* Q4 = (float4*)(Q) + (long)n * perSample4;
    const long i = (long)blockIdx.x * 256 + threadIdx.x;
    float4 v = Q4[i];
    v.x = fmaxf(v.x - mu, 0.f);
    v.y = fmaxf(v.y - mu, 0.f);
    v.z = fmaxf(v.z - mu, 0.f);
    v.w = fmaxf(v.w - mu, 0.f);
    Q4[i] = v;
}

// inv[n][m] = 1 / max(||column m||_2, 1e-12)
__global__ __launch_bounds__(256)
void q_col_inv_norm(const float* __restrict__ Q, float* __restrict__ inv, int L)
{
    const int n = blockIdx.y;
    const int m = blockIdx.x * 256 + threadIdx.x;
    const float* Qn = Q + (long)n * L * L;
    float s = 0.f;
    for (int l = 0; l < L; ++l) {
        float v = Qn[(long)l * L + m];
        s = fmaf(v, v, s);
    }
    inv[(long)n * L + m] = 1.0f / fmaxf(sqrtf(s), 1e-12f);
}

// In place: S = 0.5*(Q*invcol + (Q*invcol)^T) + eps*I
__global__ __launch_bounds__(256)
void q_normalize_symmetrize(float* __restrict__ Q, const float* __restrict__ inv,
                            int L)
{
    const int n = blockIdx.z;
    const int m = blockIdx.x * 16 + (threadIdx.x & 15);
    const int l = blockIdx.y * 16 + (threadIdx.x >> 4);
    if (l > m) return;
    float* Qn = Q + (long)n * L * L;
    const float* invn = inv + (long)n * L;
    const float q1 = Qn[(long)l * L + m] * invn[m];
    const float q2 = Qn[(long)m * L + l] * invn[l];
    float v = 0.5f * (q1 + q2);
    if (l == m) v = q1 + 1e-7f;
    Qn[(long)l * L + m] = v;
    Qn[(long)m * L + l] = v;
}

// r[n][l] = rowsum > 0 ? rsqrt(rowsum) : 0
__global__ __launch_bounds__(256)
void q_rowsum_rsqrt(const float* __restrict__ Q, float* __restrict__ rdiag, int L)
{
    const int n = blockIdx.y;
    const int l = blockIdx.x;
    const float* row = Q + ((long)n * L + l) * L;
    float s = 0.f;
    for (int m = threadIdx.x; m < L; m += 256) s += row[m];
    __shared__ float red[256];
    red[threadIdx.x] = s;
    __syncthreads();
    for (int st = 128; st > 0; st >>= 1) {
        if (threadIdx.x < st) red[threadIdx.x] += red[threadIdx.x + st];
        __syncthreads();
    }
    if (threadIdx.x == 0) {
        float rs = red[0];
        rdiag[(long)n * L + l] = (rs > 0.f) ? rsqrtf(rs) : 0.f;
    }
}

// A = S * r[l] * r[m]
__global__ __launch_bounds__(256)
void q_scale_adj(float* __restrict__ Q, const float* __restrict__ rdiag, int L)
{
    const int n = blockIdx.z;
    const int l = blockIdx.y;
    const int m = blockIdx.x * 256 + threadIdx.x;
    const float* rn = rdiag + (long)n * L;
    Q[((long)n * L + l) * L + m] *= rn[l] * rn[m];
}

// ---------------------------------------------------------------------------
// Host-side orchestration
// ---------------------------------------------------------------------------
extern "C" void kernel_launch(void* const* d_in, const int* in_sizes, int n_in,
                              void* d_out, int out_size, void* d_ws, size_t ws_size,
                              hipStream_t stream)
{
    (void)in_sizes; (void)n_in; (void)out_size; (void)ws_size;

    const int n = 32, L = 2048, dim = 512, proj = 128;
    const long NL = (long)n * L;               // 65536
    const long LL = (long)L * L;               // 4194304

    const float* x    = (const float*)d_in[0];  // [n,L,dim]
    const float* tmpM = (const float*)d_in[1];  // [dim,dim]
    const float* v0w  = (const float*)d_in[2];
    const float* v0b  = (const float*)d_in[3];
    const float* v1w  = (const float*)d_in[4];
    const float* v1b  = (const float*)d_in[5];
    const float* g1w  = (const float*)d_in[6];
    const float* g1b  = (const float*)d_in[7];
    const float* g2w  = (const float*)d_in[8];
    const float* g2b  = (const float*)d_in[9];
    float* out = (float*)d_out;

    // Workspace carve-up (floats)
    float* ws   = (float*)d_ws;
    float* wT   = ws;                       // [512,512]
    float* wW   = wT   + (long)dim * dim;   // [512,128]
    float* wZ   = wW   + (long)dim * proj;  // [n*L,128]
    float* wXW1 = wZ   + NL * proj;         // [n*L,512]
    float* wH   = wXW1 + NL * dim;          // [n*L,512]
    float* wHW2 = wH   + NL * dim;          // [n*L,512]
    float* wQ   = wHW2 + NL * dim;          // [n,L,L]
    float* wPart= wQ   + (long)n * LL;      // [n,1024]
    float* wMean= wPart + (long)n * 1024;   // [n]
    float* wInv = wMean + n;                // [n,L]
    float* wR   = wInv + (long)n * L;       // [n,L]

    dim3 blk(256);

    // 1) T = relu(temp @ v0_w + v0_b)          [512,512]
    wmma_gemm_f32<2,false><<<dim3(dim/TN, dim/TM, 1), blk, 0, stream>>>(
        tmpM, v0w, v0b, wT, dim, dim, dim, 0, 0, 0);

    // 2) W = T @ v1_w + v1_b                   [512,128]
    wmma_gemm_f32<1,false><<<dim3(proj/TN, dim/TM, 1), blk, 0, stream>>>(
        wT, v1w, v1b, wW, dim, proj, dim, 0, 0, 0);

    // 3) Z = lrelu(x @ W)                      [n*L,128]
    wmma_gemm_f32<3,false><<<dim3(proj/TN, NL/TM, 1), blk, 0, stream>>>(
        x, wW, nullptr, wZ, (int)NL, proj, dim, 0, 0, 0);

    // 4) XW1 = x @ gc1_w                       [n*L,512]
    wmma_gemm_f32<0,false><<<dim3(dim/TN, NL/TM, 1), blk, 0, stream>>>(
        x, g1w, nullptr, wXW1, (int)NL, dim, dim, 0, 0, 0);

    // 5) Q_b = Z_b @ Z_b^T                     [n,L,L]
    wmma_gemm_f32<0,true><<<dim3(L/TN, L/TM, n), blk, 0, stream>>>(
        wZ, wZ, nullptr, wQ, L, L, proj,
        (long)L * proj, (long)L * proj, LL);

    // 6) mean per sample (two-stage, deterministic)
    const int blocksPer = (int)(LL / (256 * 16));   // 1024
    q_sum_stage1<<<dim3(blocksPer, n), blk, 0, stream>>>(wQ, wPart, LL, blocksPer);
    q_sum_stage2<<<dim3(n), blk, 0, stream>>>(wPart, wMean, blocksPer, 1.0f / (float)LL);

    // 7) Q = relu(Q - mean)
    q_relu_sub_mean<<<dim3((unsigned)(LL / 4 / 256), n), blk, 0, stream>>>(
        wQ, wMean, LL / 4);

    // 8) column inverse norms
    q_col_inv_norm<<<dim3(L / 256, n), blk, 0, stream>>>(wQ, wInv, L);

    // 9) normalize columns + eps*I + symmetrize (in place)
    q_normalize_symmetrize<<<dim3(L / 16, L / 16, n), blk, 0, stream>>>(wQ, wInv, L);

    // 10) r = rsqrt(rowsum), 0 if rowsum<=0
    q_rowsum_rsqrt<<<dim3(L, n), blk, 0, stream>>>(wQ, wR, L);

    // 11) A = S * r_l * r_m (in place)
    q_scale_adj<<<dim3(L / 256, L, n), blk, 0, stream>>>(wQ, wR, L);

    // 12) h = relu(A @ XW1 + gc1_b)            [n,L,512] batched
    wmma_gemm_f32<2,false><<<dim3(dim/TN, L/TM, n), blk, 0, stream>>>(
        wQ, wXW1, g1b, wH, L, dim, L,
        LL, (long)L * dim, (long)L * dim);

    // 13) HW2 = h @ gc2_w                      [n*L,512]
    wmma_gemm_f32<0,false><<<dim3(dim/TN, NL/TM, 1), blk, 0, stream>>>(
        wH, g2w, nullptr, wHW2, (int)NL, dim, dim, 0, 0, 0);

    // 14) out = A @ HW2 + gc2_b                [n,L,512] batched
    wmma_gemm_f32<1,false><<<dim3(dim/TN, L/TM, n), blk, 0, stream>>>(
        wQ, wHW2, g2b, out, L, dim, L,
        LL, (long)L * dim, (long)L * dim);
}